// SVD_Solver_51711406244483
// MI455X (gfx1250) — compile-verified
//
#include <hip/hip_runtime.h>
#include <hip/hip_bf16.h>
#include <math.h>

typedef float v2f __attribute__((ext_vector_type(2)));
typedef float v8f __attribute__((ext_vector_type(8)));

#define NJ 24
#define MM 64
#define RS 65                       // padded LDS row stride (floats) -> conflict-free column reads
#define OFF_P 0                     // 72 rows (24 joints * 3 dims)
#define OFF_W (72*RS)               // 24 rows
#define OFF_Q (OFF_W + NJ*RS)       // 3 rows
#define OFF_T (OFF_Q + 3*RS)        // 6 wave tiles, 16x17 each
#define TSTR 17
#define TSZ  (16*TSTR)
#define OFF_ONE  (OFF_T + 6*TSZ)    // row of 1.0f (B "ones" column)
#define OFF_ZERO (OFF_ONE + RS)     // row of 0.0f (B padding columns)
#define LDS_TOTAL (OFF_ZERO + RS)   // 8197 floats = ~32.8 KB

__device__ inline float det3(const float M[3][3]) {
    return M[0][0]*(M[1][1]*M[2][2]-M[1][2]*M[2][1])
         - M[0][1]*(M[1][0]*M[2][2]-M[1][2]*M[2][0])
         + M[0][2]*(M[1][0]*M[2][1]-M[1][1]*M[2][0]);
}

// Kabsch rotation from 3x3 cross-covariance S (S = U Sig V^T; R = V diag(1,1,det) U^T)
__device__ void kabsch_rotation(const float S[3][3], float R[3][3]) {
    // A = S^T S (symmetric PSD), Jacobi eigendecomposition -> V
    float A[3][3], V[3][3] = {{1,0,0},{0,1,0},{0,0,1}};
    for (int i = 0; i < 3; ++i)
        for (int j = 0; j < 3; ++j)
            A[i][j] = S[0][i]*S[0][j] + S[1][i]*S[1][j] + S[2][i]*S[2][j];
    const int pq[3][2] = {{0,1},{0,2},{1,2}};
    for (int sweep = 0; sweep < 6; ++sweep) {
        for (int r = 0; r < 3; ++r) {
            int p = pq[r][0], q = pq[r][1];
            float apq = A[p][q];
            if (fabsf(apq) < 1e-20f) continue;
            float tau = (A[q][q] - A[p][p]) / (2.0f * apq);
            float tt  = copysignf(1.0f, tau) / (fabsf(tau) + sqrtf(1.0f + tau*tau));
            float c   = rsqrtf(1.0f + tt*tt);
            float s   = tt * c;
            for (int k = 0; k < 3; ++k) {       // A = J^T A J (columns)
                float akp = A[k][p], akq = A[k][q];
                A[k][p] = c*akp - s*akq;
                A[k][q] = s*akp + c*akq;
            }
            for (int k = 0; k < 3; ++k) {       // (rows)
                float apk = A[p][k], aqk = A[q][k];
                A[p][k] = c*apk - s*aqk;
                A[q][k] = s*apk + c*aqk;
            }
            for (int k = 0; k < 3; ++k) {       // accumulate V
                float vkp = V[k][p], vkq = V[k][q];
                V[k][p] = c*vkp - s*vkq;
                V[k][q] = s*vkp + c*vkq;
            }
        }
    }
    // sort eigenvalues descending; permute V columns
    float ev[3] = {A[0][0], A[1][1], A[2][2]};
    int o0 = 0, o1 = 1, o2 = 2, t;
    if (ev[o0] < ev[o1]) { t = o0; o0 = o1; o1 = t; }
    if (ev[o0] < ev[o2]) { t = o0; o0 = o2; o2 = t; }
    if (ev[o1] < ev[o2]) { t = o1; o1 = o2; o2 = t; }
    float Vc[3][3];
    for (int k = 0; k < 3; ++k) { Vc[k][0] = V[k][o0]; Vc[k][1] = V[k][o1]; Vc[k][2] = V[k][o2]; }
    // U columns: u_i = S v_i (orthonormalized); u2 via cross, sign matched to S v2
    float u0[3], u1[3], u2[3], w2[3];
    for (int k = 0; k < 3; ++k) {
        u0[k] = S[k][0]*Vc[0][0] + S[k][1]*Vc[1][0] + S[k][2]*Vc[2][0];
        u1[k] = S[k][0]*Vc[0][1] + S[k][1]*Vc[1][1] + S[k][2]*Vc[2][1];
        w2[k] = S[k][0]*Vc[0][2] + S[k][1]*Vc[1][2] + S[k][2]*Vc[2][2];
    }
    float n0 = rsqrtf(u0[0]*u0[0] + u0[1]*u0[1] + u0[2]*u0[2] + 1e-30f);
    for (int k = 0; k < 3; ++k) u0[k] *= n0;
    float d01 = u1[0]*u0[0] + u1[1]*u0[1] + u1[2]*u0[2];
    for (int k = 0; k < 3; ++k) u1[k] -= d01 * u0[k];
    float n1 = rsqrtf(u1[0]*u1[0] + u1[1]*u1[1] + u1[2]*u1[2] + 1e-30f);
    for (int k = 0; k < 3; ++k) u1[k] *= n1;
    u2[0] = u0[1]*u1[2] - u0[2]*u1[1];
    u2[1] = u0[2]*u1[0] - u0[0]*u1[2];
    u2[2] = u0[0]*u1[1] - u0[1]*u1[0];
    if (u2[0]*w2[0] + u2[1]*w2[1] + u2[2]*w2[2] < 0.0f)
        for (int k = 0; k < 3; ++k) u2[k] = -u2[k];
    float U[3][3];
    for (int k = 0; k < 3; ++k) { U[k][0] = u0[k]; U[k][1] = u1[k]; U[k][2] = u2[k]; }
    float d = (det3(U) * det3(Vc) < 0.0f) ? -1.0f : 1.0f;   // det(V U^T)
    for (int i = 0; i < 3; ++i)
        for (int j = 0; j < 3; ++j)
            R[i][j] = Vc[i][0]*U[j][0] + Vc[i][1]*U[j][1] + d*Vc[i][2]*U[j][2];
}

__global__ __launch_bounds__(192)
void kabsch_wmma_kernel(const float* __restrict__ points,   // (B, 64, 3)
                        const float* __restrict__ weight,   // (B, 64, 24)
                        const float* __restrict__ offset,   // (B, 64, 24, 3)
                        float* __restrict__ out, int B)
{
    __shared__ float lds[LDS_TOTAL];
    const int b   = blockIdx.x;
    const int tid = threadIdx.x;

    // constant rows for the B matrix (ones column, zero padding columns)
    if (tid < RS) {
        lds[OFF_ONE  + tid] = 1.0f;
        lds[OFF_ZERO + tid] = 0.0f;
    }

    // ---- stage batch into LDS with transpose (coalesced b128 global reads) ----
    const float4* po = (const float4*)(offset + (size_t)b * (MM*NJ*3));
    for (int i = tid; i < (MM*NJ*3)/4; i += 192) {
        float4 v = po[i];
        float vals[4] = {v.x, v.y, v.z, v.w};
        int base = i * 4;
        #pragma unroll
        for (int e = 0; e < 4; ++e) {
            int flat = base + e;
            int m = flat / 72, r = flat - m * 72;        // r = j*3 + d
            lds[OFF_P + r * RS + m] = vals[e];
        }
    }
    const float4* pw = (const float4*)(weight + (size_t)b * (MM*NJ));
    for (int i = tid; i < (MM*NJ)/4; i += 192) {
        float4 v = pw[i];
        float vals[4] = {v.x, v.y, v.z, v.w};
        int base = i * 4;
        #pragma unroll
        for (int e = 0; e < 4; ++e) {
            int flat = base + e;
            int m = flat / 24, j = flat - m * 24;
            lds[OFF_W + j * RS + m] = vals[e];
        }
    }
    if (tid < (MM*3)/4) {
        float4 v = ((const float4*)(points + (size_t)b * (MM*3)))[tid];
        float vals[4] = {v.x, v.y, v.z, v.w};
        int base = tid * 4;
        #pragma unroll
        for (int e = 0; e < 4; ++e) {
            int flat = base + e;
            int m = flat / 3, d = flat - m * 3;
            lds[OFF_Q + d * RS + m] = vals[e];
        }
    }
    __syncthreads();

    // ---- fused reductions via V_WMMA_F32_16X16X4_F32, 4 joints per wave ----
    const int wave = tid >> 5, lane = tid & 31;
    const int hi = lane >> 4, lo = lane & 15;
    const int j0 = wave * 4;

    // A row lo: joint j0+lo/4, component lo%4 (x,y,z of P, or w)
    const int jjA = j0 + (lo >> 2), cA = lo & 3;
    const int aoff = (cA < 3) ? (OFF_P + (jjA * 3 + cA) * RS) : (OFF_W + jjA * RS);

    // B col lo: Q[x,y,z], ones, w_{j0..j0+3}, zeros — all unconditional LDS rows
    int boff;
    if (lo < 3)       boff = OFF_Q + lo * RS;
    else if (lo == 3) boff = OFF_ONE;
    else if (lo < 8)  boff = OFF_W + (j0 + lo - 4) * RS;
    else              boff = OFF_ZERO;       // same addr across lanes -> LDS broadcast

    v8f acc = {};
    const int koff = hi * 2;
    #pragma unroll
    for (int kt = 0; kt < 16; ++kt) {
        int m0 = kt * 4 + koff;
        v2f a, bb;
        a[0]  = lds[aoff + m0];
        a[1]  = lds[aoff + m0 + 1];
        bb[0] = lds[boff + m0];
        bb[1] = lds[boff + m0 + 1];
        acc = __builtin_amdgcn_wmma_f32_16x16x4_f32(
                /*neg_a=*/false, a, /*neg_b=*/false, bb,
                /*c_mod=*/(short)0, acc, /*reuse_a=*/false, /*reuse_b=*/false);
    }

    // dump 16x16 fragment to this wave's padded LDS tile
    const int tbase = OFF_T + wave * TSZ;
    #pragma unroll
    for (int v = 0; v < 8; ++v)
        lds[tbase + (v + 8 * hi) * TSTR + lo] = acc[v];
    __syncthreads();

    // ---- per-joint 3x3 assembly + SVD (lanes 0..3 of each wave) ----
    if (lane < 4) {
        const int jj = j0 + lane;
        const int rb = lane * 4;
        float PQ[3][3], sumP[3], wP[3], wQ[3];
        #pragma unroll
        for (int c = 0; c < 3; ++c) {
            #pragma unroll
            for (int e = 0; e < 3; ++e) PQ[c][e] = lds[tbase + (rb + c) * TSTR + e];
            sumP[c] = lds[tbase + (rb + c) * TSTR + 3];
            wP[c]   = lds[tbase + (rb + c) * TSTR + 4 + lane];
        }
        #pragma unroll
        for (int e = 0; e < 3; ++e) wQ[e] = lds[tbase + (rb + 3) * TSTR + e];
        float wsum = lds[tbase + (rb + 3) * TSTR + 3];

        float sumQ[3];
        #pragma unroll
        for (int e = 0; e < 3; ++e) {
            float s = 0.0f;
            for (int m = 0; m < MM; ++m) s += lds[OFF_Q + e * RS + m];
            sumQ[e] = s;
        }

        const float inv = 1.0f / wsum;
        float Pc[3], Qc[3];
        #pragma unroll
        for (int k = 0; k < 3; ++k) { Pc[k] = wP[k] * inv; Qc[k] = wQ[k] * inv; }

        // S = PQ - Pc*sumQ^T - sumP*Qc^T + M*Pc*Qc^T
        float S[3][3];
        #pragma unroll
        for (int c = 0; c < 3; ++c)
            #pragma unroll
            for (int e = 0; e < 3; ++e)
                S[c][e] = PQ[c][e] - Pc[c]*sumQ[e] - Qc[e]*sumP[c] + (float)MM * Pc[c]*Qc[e];

        float R[3][3];
        kabsch_rotation(S, R);

        float tv[3];
        #pragma unroll
        for (int i = 0; i < 3; ++i)
            tv[i] = Qc[i] - (R[i][0]*Pc[0] + R[i][1]*Pc[1] + R[i][2]*Pc[2]);

        const size_t rbase = ((size_t)b * NJ + jj) * 9;
        #pragma unroll
        for (int i = 0; i < 3; ++i)
            #pragma unroll
            for (int j = 0; j < 3; ++j)
                out[rbase + i * 3 + j] = R[i][j];
        const size_t tb = (size_t)B * NJ * 9 + ((size_t)b * NJ + jj) * 3;
        #pragma unroll
        for (int i = 0; i < 3; ++i) out[tb + i] = tv[i];
    }
}

extern "C" void kernel_launch(void* const* d_in, const int* in_sizes, int n_in,
                              void* d_out, int out_size, void* d_ws, size_t ws_size,
                              hipStream_t stream) {
    const float* points = (const float*)d_in[0];
    const float* weight = (const float*)d_in[1];
    const float* offset = (const float*)d_in[2];
    float* out = (float*)d_out;
    const int B = in_sizes[0] / (MM * 3);   // 4096
    kabsch_wmma_kernel<<<B, 192, 0, stream>>>(points, weight, offset, out, B);
}